// GaussianModel_65197603554117
// MI455X (gfx1250) — compile-verified
//
#include <hip/hip_runtime.h>

typedef float v2f __attribute__((ext_vector_type(2)));
typedef float v8f __attribute__((ext_vector_type(8)));

#define P_N 8192
#define G_N 4096
#define GTILES (G_N / 16)              // 256 gaussian tiles of 16
#define GSPLIT 16                      // gaussian-range splits (grid.y)
#define WAVES_PER_BLOCK 8
#define TILE_F2 384                    // v2f elements per gtile: 4 quantities * 3 K-chunks * 32 lanes
#define RANGE_TILES (GTILES / GSPLIT)  // 16 tiles per block
#define TILE_BYTES (TILE_F2 * 8)       // 3072 B per tile
#define COEF_BYTES (RANGE_TILES * TILE_BYTES)           // 49152 B
#define SMEM_BYTES (COEF_BYTES + RANGE_TILES * 16 * 4)  // + 1 KB c0

// ---------------------------------------------------------------------------
// Kernel 1: per-gaussian coefficient precompute, written in WMMA-B layout.
// Quantities: 0 = exp2-arg (-0.5*log2e*mahal), 1 = |diff|^2, 2 = alpha*SH-linear,
//             3 = alpha*SH-quadratic.  K basis: [x2,y2,z2,xy,xz,yz,x,y,z,1,0,0]
// ---------------------------------------------------------------------------
__global__ void gs_precompute(const float* __restrict__ mean,
                              const float* __restrict__ fdc,
                              const float* __restrict__ frest,
                              const float* __restrict__ scaling,
                              const float* __restrict__ rotation,
                              const float* __restrict__ opacity,
                              float* __restrict__ c0out,
                              float* __restrict__ cfout) {
  int g = blockIdx.x * blockDim.x + threadIdx.x;
  if (g >= G_N) return;

  const float C0f = 0.28209479177387814f;
  const float C1f = 0.4886025119029199f;
  const float C2_0 = 1.0925484305920792f;
  const float C2_1 = -1.0925484305920792f;
  const float C2_2 = 0.31539156525252005f;
  const float C2_3 = -1.0925484305920792f;
  const float C2_4 = 0.5462742152960396f;

  float mx = mean[g * 3 + 0], my = mean[g * 3 + 1], mz = mean[g * 3 + 2];

  // quaternion -> rotation
  float qr = rotation[g * 4 + 0], qx = rotation[g * 4 + 1];
  float qy = rotation[g * 4 + 2], qz = rotation[g * 4 + 3];
  float qn = rsqrtf(qr * qr + qx * qx + qy * qy + qz * qz);
  qr *= qn; qx *= qn; qy *= qn; qz *= qn;
  float R00 = 1.f - 2.f * (qy * qy + qz * qz), R01 = 2.f * (qx * qy - qr * qz), R02 = 2.f * (qx * qz + qr * qy);
  float R10 = 2.f * (qx * qy + qr * qz), R11 = 1.f - 2.f * (qx * qx + qz * qz), R12 = 2.f * (qy * qz - qr * qx);
  float R20 = 2.f * (qx * qz - qr * qy), R21 = 2.f * (qy * qz + qr * qx), R22 = 1.f - 2.f * (qx * qx + qy * qy);

  // inv_var = exp(-2*scaling); Sinv = R diag(iv) R^T (symmetric)
  float iv0 = expf(-2.f * scaling[g * 3 + 0]);
  float iv1 = expf(-2.f * scaling[g * 3 + 1]);
  float iv2 = expf(-2.f * scaling[g * 3 + 2]);
  float S00 = R00 * R00 * iv0 + R01 * R01 * iv1 + R02 * R02 * iv2;
  float S11 = R10 * R10 * iv0 + R11 * R11 * iv1 + R12 * R12 * iv2;
  float S22 = R20 * R20 * iv0 + R21 * R21 * iv1 + R22 * R22 * iv2;
  float S01 = R00 * R10 * iv0 + R01 * R11 * iv1 + R02 * R12 * iv2;
  float S02 = R00 * R20 * iv0 + R01 * R21 * iv1 + R02 * R22 * iv2;
  float S12 = R10 * R20 * iv0 + R11 * R21 * iv1 + R12 * R22 * iv2;
  float u0 = S00 * mx + S01 * my + S02 * mz;
  float u1 = S01 * mx + S11 * my + S12 * mz;
  float u2 = S02 * mx + S12 * my + S22 * mz;

  float alpha = 1.f / (1.f + expf(-opacity[g]));
  float sh0 = fdc[g];
  float s1 = frest[g * 8 + 0], s2 = frest[g * 8 + 1], s3 = frest[g * 8 + 2];
  float s4 = frest[g * 8 + 3], s5 = frest[g * 8 + 4], s6 = frest[g * 8 + 5];
  float s7 = frest[g * 8 + 6], s8 = frest[g * 8 + 7];

  float c[4][12];
#pragma unroll
  for (int i = 0; i < 4; ++i)
#pragma unroll
    for (int k = 0; k < 12; ++k) c[i][k] = 0.f;

  // qt=0: exp2 argument = -0.5*log2(e) * mahal(q)
  const float kE = -0.5f * 1.4426950408889634f;
  c[0][0] = kE * S00; c[0][1] = kE * S11; c[0][2] = kE * S22;
  c[0][3] = kE * 2.f * S01; c[0][4] = kE * 2.f * S02; c[0][5] = kE * 2.f * S12;
  c[0][6] = kE * -2.f * u0; c[0][7] = kE * -2.f * u1; c[0][8] = kE * -2.f * u2;
  c[0][9] = kE * (mx * u0 + my * u1 + mz * u2);

  // qt=1: |q - m|^2
  c[1][0] = 1.f; c[1][1] = 1.f; c[1][2] = 1.f;
  c[1][6] = -2.f * mx; c[1][7] = -2.f * my; c[1][8] = -2.f * mz;
  c[1][9] = mx * mx + my * my + mz * mz;

  // qt=2: alpha * (a . diff), a = (-C1*sh3, -C1*sh1, C1*sh2)
  float a0 = -C1f * s3, a1 = -C1f * s1, a2 = C1f * s2;
  c[2][6] = alpha * a0; c[2][7] = alpha * a1; c[2][8] = alpha * a2;
  c[2][9] = alpha * -(a0 * mx + a1 * my + a2 * mz);

  // qt=3: alpha * diff^T B diff
  float B00 = -C2_2 * s6 + C2_4 * s8;
  float B11 = -C2_2 * s6 - C2_4 * s8;
  float B22 = 2.f * C2_2 * s6;
  float B01 = 0.5f * C2_0 * s4;
  float B02 = 0.5f * C2_3 * s7;
  float B12 = 0.5f * C2_1 * s5;
  float v0 = B00 * mx + B01 * my + B02 * mz;
  float v1 = B01 * mx + B11 * my + B12 * mz;
  float v2 = B02 * mx + B12 * my + B22 * mz;
  c[3][0] = alpha * B00; c[3][1] = alpha * B11; c[3][2] = alpha * B22;
  c[3][3] = alpha * 2.f * B01; c[3][4] = alpha * 2.f * B02; c[3][5] = alpha * 2.f * B12;
  c[3][6] = alpha * -2.f * v0; c[3][7] = alpha * -2.f * v1; c[3][8] = alpha * -2.f * v2;
  c[3][9] = alpha * (mx * v0 + my * v1 + mz * v2);

  c0out[g] = alpha * C0f * sh0;

  // scatter into WMMA-B layout: lane L of a wave loads v2f at
  // [gtile*384 + qt*96 + chunk*32 + L], giving (K pair, N=column) per the
  // 32-bit 4x16 B-matrix layout (lanes 0-15: K0,K1; lanes 16-31: K2,K3).
  int t = g >> 4, gi = g & 15;
#pragma unroll
  for (int qt = 0; qt < 4; ++qt)
#pragma unroll
    for (int k = 0; k < 12; ++k) {
      int ch = k >> 2, h = (k >> 1) & 1, kk = k & 1;
      size_t idx = ((size_t)t * TILE_F2 + qt * 96 + ch * 32 + h * 16 + gi) * 2 + kk;
      cfout[idx] = c[qt][k];
    }
}

// ---------------------------------------------------------------------------
// Kernel 2: zero output (harness poisons d_out; we accumulate with atomics)
// ---------------------------------------------------------------------------
__global__ void gs_zero(float* __restrict__ out) {
  int i = blockIdx.x * blockDim.x + threadIdx.x;
  if (i < P_N) out[i] = 0.f;
}

// ---------------------------------------------------------------------------
// Kernel 3: async-stage B range into LDS once, then tiled 16x16 WMMA GEMM
// (f32 16x16x4, K=12) + transcendental epilogue fed from ds_load.
// ---------------------------------------------------------------------------
__global__ __launch_bounds__(256)
void gs_main(const float* __restrict__ pos,
             const float* __restrict__ c0arr,
             const v2f* __restrict__ cf,
             float* __restrict__ out) {
  __shared__ char smem[SMEM_BYTES];

  const int tid = threadIdx.x;
  const int lane = tid & 31;
  const int wave = tid >> 5;
  const int ptile = blockIdx.x * WAVES_PER_BLOCK + wave;
  const int prow0 = ptile * 16;
  const int p_local = lane & 15;
  const int p = prow0 + p_local;
  const int t0 = blockIdx.y * RANGE_TILES;

  // ---- stage this block's 48KB coefficient range + 1KB c0 into LDS via
  //      GLOBAL_LOAD_ASYNC_TO_LDS_B128 (ASYNCcnt), one wait + one barrier ----
  {
    const char* gbase = (const char*)(cf + (size_t)t0 * TILE_F2);
    const unsigned lbase = (unsigned)(uintptr_t)&smem[0];
#pragma unroll
    for (int k = 0; k < COEF_BYTES / (256 * 16); ++k) {  // 12 x b128 per thread
      const int idx = tid + k * 256;
      const unsigned loff = lbase + idx * 16;
      const char* gptr = gbase + (size_t)idx * 16;
      asm volatile("global_load_async_to_lds_b128 %0, %1, off"
                   :: "v"(loff), "v"(gptr) : "memory");
    }
    {  // c0 for the 16 tiles: 256 floats, one b32 per thread
      const unsigned loff = lbase + COEF_BYTES + tid * 4;
      const char* gptr = (const char*)(c0arr + t0 * 16 + tid);
      asm volatile("global_load_async_to_lds_b32 %0, %1, off"
                   :: "v"(loff), "v"(gptr) : "memory");
    }
    asm volatile("s_wait_asynccnt 0" ::: "memory");
    __syncthreads();
  }

  const v2f* sB = (const v2f*)smem;
  const float* sC0 = (const float*)(smem + COEF_BYTES);

  // point features f_p (A-matrix rows). A 16x4 f32 layout: lanes 0-15 hold
  // (K0,K1) of row M=lane; lanes 16-31 hold (K2,K3) of row M=lane-16.
  float x = pos[p * 3 + 0] * 2.f - 1.f;
  float y = pos[p * 3 + 1] * 2.f - 1.f;
  float z = pos[p * 3 + 2] * 2.f - 1.f;
  float f[12] = { x * x, y * y, z * z, x * y, x * z, y * z, x, y, z, 1.f, 0.f, 0.f };
  const int koff = (lane >> 4) * 2;
  v2f A0 = { f[koff + 0], f[koff + 1] };
  v2f A1 = { f[koff + 4], f[koff + 5] };
  v2f A2 = { f[koff + 8], f[koff + 9] };

  float acc[8];
#pragma unroll
  for (int j = 0; j < 8; ++j) acc[j] = 0.f;

  for (int tl = 0; tl < RANGE_TILES; ++tl) {
    const v2f* bp = sB + tl * TILE_F2 + lane;  // ds_load_b64 at static offsets

    v8f E = {0.f, 0.f, 0.f, 0.f, 0.f, 0.f, 0.f, 0.f};
    v8f D = E, L = E, Q = E;

    // 4 accumulator chains x 3 K-chunks of V_WMMA_F32_16X16X4_F32
    E = __builtin_amdgcn_wmma_f32_16x16x4_f32(false, A0, false, bp[0],   (short)0, E, false, false);
    E = __builtin_amdgcn_wmma_f32_16x16x4_f32(false, A1, false, bp[32],  (short)0, E, false, false);
    E = __builtin_amdgcn_wmma_f32_16x16x4_f32(false, A2, false, bp[64],  (short)0, E, false, false);
    D = __builtin_amdgcn_wmma_f32_16x16x4_f32(false, A0, false, bp[96],  (short)0, D, false, false);
    D = __builtin_amdgcn_wmma_f32_16x16x4_f32(false, A1, false, bp[128], (short)0, D, false, false);
    D = __builtin_amdgcn_wmma_f32_16x16x4_f32(false, A2, false, bp[160], (short)0, D, false, false);
    L = __builtin_amdgcn_wmma_f32_16x16x4_f32(false, A0, false, bp[192], (short)0, L, false, false);
    L = __builtin_amdgcn_wmma_f32_16x16x4_f32(false, A1, false, bp[224], (short)0, L, false, false);
    L = __builtin_amdgcn_wmma_f32_16x16x4_f32(false, A2, false, bp[256], (short)0, L, false, false);
    Q = __builtin_amdgcn_wmma_f32_16x16x4_f32(false, A0, false, bp[288], (short)0, Q, false, false);
    Q = __builtin_amdgcn_wmma_f32_16x16x4_f32(false, A1, false, bp[320], (short)0, Q, false, false);
    Q = __builtin_amdgcn_wmma_f32_16x16x4_f32(false, A2, false, bp[352], (short)0, Q, false, false);

    // epilogue, lane-local on the C layout: lane's column g = (t0+tl)*16 + (lane&15)
    const float c0l = sC0[tl * 16 + p_local];
#pragma unroll
    for (int j = 0; j < 8; ++j) {
      float gsv = __builtin_amdgcn_exp2f(E[j]);                 // gauss = exp(-.5*mahal)
      float s   = __builtin_amdgcn_sqrtf(D[j]);                 // |diff|
      float inv = __builtin_amdgcn_rcpf(s + 1e-8f);             // 1/(|diff|+eps)
      float feat = fmaf(inv * inv, Q[j], fmaf(inv, L[j], c0l)); // alpha*eval_sh
      acc[j] = fmaf(gsv, feat, acc[j]);
    }
  }

  // sum over the 16 columns held across lanes of each half-wave
#pragma unroll
  for (int m = 1; m < 16; m <<= 1)
#pragma unroll
    for (int j = 0; j < 8; ++j) acc[j] += __shfl_xor(acc[j], m, 32);

  if (p_local == 0) {
    const int base = prow0 + (lane >> 4) * 8;  // rows M=j (lanes<16) / M=j+8 (lanes>=16)
#pragma unroll
    for (int j = 0; j < 8; ++j) atomicAdd(out + base + j, acc[j]);
  }
}

// ---------------------------------------------------------------------------
extern "C" void kernel_launch(void* const* d_in, const int* in_sizes, int n_in,
                              void* d_out, int out_size, void* d_ws, size_t ws_size,
                              hipStream_t stream) {
  const float* pos      = (const float*)d_in[0];  // (P,3)
  const float* mean     = (const float*)d_in[1];  // (G,3)
  const float* fdc      = (const float*)d_in[2];  // (G,1,1)
  const float* frest    = (const float*)d_in[3];  // (G,1,8)
  const float* scaling  = (const float*)d_in[4];  // (G,3)
  const float* rotation = (const float*)d_in[5];  // (G,4)
  const float* opacity  = (const float*)d_in[6];  // (G,1)
  float* out = (float*)d_out;

  float* c0 = (float*)d_ws;                       // G floats
  float* cfraw = c0 + G_N;                        // 256 tiles * 384 v2f = 768 KB
  (void)in_sizes; (void)n_in; (void)out_size; (void)ws_size;

  gs_precompute<<<G_N / 256, 256, 0, stream>>>(mean, fdc, frest, scaling, rotation,
                                               opacity, c0, cfraw);
  gs_zero<<<P_N / 256, 256, 0, stream>>>(out);

  dim3 grid(P_N / (16 * WAVES_PER_BLOCK), GSPLIT);  // (64, 16) -> 1024 blocks
  gs_main<<<grid, 256, 0, stream>>>(pos, c0, (const v2f*)cfraw, out);
}